// FCOS_39659728011713
// MI455X (gfx1250) — compile-verified
//
#include <hip/hip_runtime.h>
#include <hip/hip_bf16.h>
#include <stdint.h>

#define NCLS      80
#define NPTS      21824
#define NBINS     4096
#define CAND_CAP  4096
#define PRE_TOPK  1000
#define POST_TOPK 100
#define NIMG      16
#define NMS_T     0.6f
#define CLS_OFF   4096.0f

// ---------------- workspace layout (bytes) ----------------
enum : size_t {
  OFF_HIST   = 0,                                   // 16*4096 ints = 256KB
  OFF_THRESH = OFF_HIST + (size_t)NIMG * NBINS * 4, // 16 ints (64B slot)
  OFF_CNT    = OFF_THRESH + 64,
  OFF_KCNT   = OFF_CNT + 64,
  OFF_CAND   = OFF_KCNT + 64,                       // 16*4096 uint2 = 512KB
  OFF_SSC    = OFF_CAND + (size_t)NIMG * CAND_CAP * 8,
  OFF_SID    = OFF_SSC + (size_t)NIMG * 1024 * 4,
  OFF_SX1    = OFF_SID + (size_t)NIMG * 1024 * 4,
  OFF_SY1    = OFF_SX1 + (size_t)NIMG * 1024 * 4,
  OFF_SX2    = OFF_SY1 + (size_t)NIMG * 1024 * 4,
  OFF_SY2    = OFF_SX2 + (size_t)NIMG * 1024 * 4,
  OFF_SAR    = OFF_SY2 + (size_t)NIMG * 1024 * 4,
  OFF_MASK   = OFF_SAR + (size_t)NIMG * 1024 * 4,   // 16*1024*32 words = 2MB
  WS_TOTAL   = OFF_MASK + (size_t)NIMG * 1024 * 32 * 4
};

struct Ptrs {
  const float* cls[5];
  const float* box[5];
  const float* ctr[5];
};

typedef unsigned u32x4 __attribute__((ext_vector_type(4)));
typedef unsigned u32x8 __attribute__((ext_vector_type(8)));

__device__ __forceinline__ float sigmoidf(float x) { return 1.0f / (1.0f + __expf(-x)); }

__device__ __forceinline__ void lvl_info(int pt, int& lvl, int& off, int& W, int& HW, float& s) {
  if      (pt < 16384) { lvl = 0; off = pt;         W = 128; HW = 16384; s = 8.f;   }
  else if (pt < 20480) { lvl = 1; off = pt - 16384; W = 64;  HW = 4096;  s = 16.f;  }
  else if (pt < 21504) { lvl = 2; off = pt - 20480; W = 32;  HW = 1024;  s = 32.f;  }
  else if (pt < 21760) { lvl = 3; off = pt - 21504; W = 16;  HW = 256;   s = 64.f;  }
  else                 { lvl = 4; off = pt - 21760; W = 8;   HW = 64;    s = 128.f; }
}

__device__ __forceinline__ void decode_box(const Ptrs& p, int img, int loc,
                                           float& x1, float& y1, float& x2, float& y2) {
  int lvl, off, W, HW; float s;
  lvl_info(loc, lvl, off, W, HW, s);
  int h = off / W, w = off - h * W;
  float px = (float)w * s + 0.5f * s;
  float py = (float)h * s + 0.5f * s;
  const float* bb = p.box[lvl] + (size_t)img * 4 * HW + off;
  float b0 = bb[0], b1 = bb[HW], b2 = bb[2 * HW], b3 = bb[3 * HW];
  x1 = px - b0 * s; y1 = py - b1 * s; x2 = px + b2 * s; y2 = py + b3 * s;
}

// ---- CDNA5 async global->LDS copy (ASYNCcnt path), ISA §15.18 opcode 98 ----
__device__ __forceinline__ void async_g2lds_b128(unsigned lds_off, const void* g) {
  asm volatile("global_load_async_to_lds_b128 %0, %1, off"
               :: "v"(lds_off), "v"((unsigned long long)(uintptr_t)g)
               : "memory");
}
__device__ __forceinline__ void wait_asynccnt0() {
  asm volatile("s_wait_asynccnt 0" ::: "memory");
}

// ---- CDNA5 Tensor Data Mover: 1-line tile of `ndw` dwords -> LDS ------------
// D# per ISA cdna5 §8.3/8.4: group0 {count=1, lds_addr, global_addr, type=2},
// group1 {wg_mask=0, data_size=4B, tensor_dim0=tile_dim0=ndw, dim0_stride=ndw}.
__device__ __forceinline__ void tdm_load_1d(unsigned lds_off, const void* g, unsigned ndw) {
  unsigned long long ga = (unsigned long long)(uintptr_t)g;
  u32x4 g0;
  g0[0] = 1u;                                            // count=1, user desc
  g0[1] = lds_off;                                       // lds_addr (bytes)
  g0[2] = (unsigned)ga;                                  // global_addr[31:0]
  g0[3] = (unsigned)((ga >> 32) & 0x1ffffffu) | (2u << 30); // addr[56:32] | type=2
  u32x8 g1;
  g1[0] = (2u << 16);                                    // data_size=4B, wg_mask=0
  g1[1] = (ndw & 0xffffu) << 16;                         // tensor_dim0[15:0]
  g1[2] = ((ndw >> 16) & 0xffffu) | (1u << 16);          // tensor_dim0[31:16] | tensor_dim1=1
  g1[3] = (ndw & 0xffffu) << 16;                         // tile_dim0
  g1[4] = 0u;                                            // tile_dim1/2 unused
  g1[5] = ndw;                                           // tensor_dim0_stride[31:0]
  g1[6] = 0u;
  g1[7] = 0u;
  asm volatile("tensor_load_to_lds %0, %1"
               :: "s"(g0), "s"(g1)
               : "memory");
}

// ---------------- K0: zero counters/histograms ----------------
__global__ void k_init(int* __restrict__ w, int n) {
  int i = blockIdx.x * blockDim.x + threadIdx.x;
  if (i < n) w[i] = 0;
}

// ---------------- K1: fused score + per-block LDS histogram ----------------
__global__ void __launch_bounds__(256) k_hist(Ptrs p, int* __restrict__ hist) {
  __shared__ int h[NBINS];
  for (int i = threadIdx.x; i < NBINS; i += blockDim.x) h[i] = 0;
  __syncthreads();
  const int img = blockIdx.y;
  const int per = NCLS * NPTS;
  const int stride = gridDim.x * blockDim.x;
  for (int e = blockIdx.x * blockDim.x + threadIdx.x; e < per; e += stride) {
    int c = e / NPTS;
    int pt = e - c * NPTS;
    int lvl, off, W, HW; float sfl;
    lvl_info(pt, lvl, off, W, HW, sfl);
    const float* cp = p.cls[lvl] + (size_t)(img * NCLS + c) * HW + off;
    __builtin_prefetch(cp + 2048, 0, 1);   // gfx1250 global_prefetch_b8
    float sc = sigmoidf(*cp);
    if (sc > 0.05f) {
      float ct = sigmoidf(p.ctr[lvl][(size_t)img * HW + off]);
      float comb = sc * ct;
      int bin = (int)(comb * (float)NBINS);
      bin = bin < 0 ? 0 : (bin > NBINS - 1 ? NBINS - 1 : bin);
      atomicAdd(&h[bin], 1);
    }
  }
  __syncthreads();
  for (int i = threadIdx.x; i < NBINS; i += blockDim.x)
    if (h[i] != 0) atomicAdd(&hist[img * NBINS + i], h[i]);
}

// ---------------- K2: per-image threshold bin (top-1000 superset) ----------------
__global__ void k_thresh(const int* __restrict__ hist, int* __restrict__ thresh) {
  int img = threadIdx.x;
  if (img < NIMG) {
    int acc = 0, t = 0;
    for (int b = NBINS - 1; b >= 0; --b) {
      acc += hist[img * NBINS + b];
      if (acc >= PRE_TOPK) { t = b; break; }
    }
    thresh[img] = t;
  }
}

// ---------------- K3: compact candidates above threshold ----------------
__global__ void __launch_bounds__(256) k_compact(Ptrs p, const int* __restrict__ thresh,
                                                 int* __restrict__ cnt, uint2* __restrict__ cand) {
  const int img = blockIdx.y;
  const int tb = thresh[img];
  const int per = NCLS * NPTS;
  const int stride = gridDim.x * blockDim.x;
  for (int e = blockIdx.x * blockDim.x + threadIdx.x; e < per; e += stride) {
    int c = e / NPTS;
    int pt = e - c * NPTS;
    int lvl, off, W, HW; float sfl;
    lvl_info(pt, lvl, off, W, HW, sfl);
    float sc = sigmoidf(p.cls[lvl][(size_t)(img * NCLS + c) * HW + off]);
    if (sc > 0.05f) {
      float comb = sc * sigmoidf(p.ctr[lvl][(size_t)img * HW + off]);
      int bin = (int)(comb * (float)NBINS);
      bin = bin < 0 ? 0 : (bin > NBINS - 1 ? NBINS - 1 : bin);
      if (bin >= tb) {
        int pos = atomicAdd(&cnt[img], 1);
        if (pos < CAND_CAP)
          cand[(size_t)img * CAND_CAP + pos] =
              make_uint2(__float_as_uint(comb), (unsigned)(pt * NCLS + c));
      }
    }
  }
}

// ---------------- K4: async-DMA candidates to LDS, bitonic sort, decode ----------------
__global__ void __launch_bounds__(1024) k_sort_decode(Ptrs p,
    const int* __restrict__ cnt, const uint2* __restrict__ cand, int* __restrict__ kcnt,
    float* __restrict__ ssc, unsigned* __restrict__ sid,
    float* __restrict__ sx1, float* __restrict__ sy1,
    float* __restrict__ sx2, float* __restrict__ sy2, float* __restrict__ sar) {
  __shared__ uint2 s[CAND_CAP];   // 32 KB
  const int img = blockIdx.x;
  const int tid = threadIdx.x;
  const int NT = blockDim.x;
  int n = cnt[img]; if (n > CAND_CAP) n = CAND_CAP;

  const char* src = (const char*)(cand + (size_t)img * CAND_CAP);
  unsigned lbase = (unsigned)(uintptr_t)(&s[0]);
  for (int c = tid; c < CAND_CAP * 8 / 16; c += NT)
    async_g2lds_b128(lbase + c * 16, src + (size_t)c * 16);
  wait_asynccnt0();
  __syncthreads();

  for (int i = n + tid; i < CAND_CAP; i += NT) { s[i].x = 0u; s[i].y = 0x7fffffffu; }
  __syncthreads();

  // bitonic sort, descending by score bits (positive floats), idx-ascending tiebreak
  for (int k = 2; k <= CAND_CAP; k <<= 1) {
    for (int j = k >> 1; j > 0; j >>= 1) {
      for (int i = tid; i < CAND_CAP; i += NT) {
        int ixj = i ^ j;
        if (ixj > i) {
          uint2 a = s[i], b = s[ixj];
          bool aFirst = (a.x > b.x) || (a.x == b.x && a.y < b.y);
          bool descSeg = ((i & k) == 0);
          if (descSeg ? !aFirst : aFirst) { s[i] = b; s[ixj] = a; }
        }
      }
      __syncthreads();
    }
  }

  int K = n < PRE_TOPK ? n : PRE_TOPK;
  if (tid == 0) kcnt[img] = K;

  for (int i = tid; i < 1024; i += NT) {
    float score = -1.0f; unsigned idx = 0u;
    float x1 = 0.f, y1 = 0.f, x2 = 0.f, y2 = 0.f, ar = 0.f;
    if (i < K) {
      uint2 v = s[i];
      score = __uint_as_float(v.x);
      idx = v.y;
      int loc = (int)idx / NCLS;
      int c = (int)idx - loc * NCLS;
      decode_box(p, img, loc, x1, y1, x2, y2);
      float o = (float)c * CLS_OFF;                   // class-offset boxes for NMS
      x1 += o; y1 += o; x2 += o; y2 += o;
      ar = fmaxf(x2 - x1, 0.f) * fmaxf(y2 - y1, 0.f);
    }
    size_t d = (size_t)img * 1024 + i;
    ssc[d] = score; sid[d] = idx;
    sx1[d] = x1; sy1[d] = y1; sx2[d] = x2; sy2[d] = y2; sar[d] = ar;
  }
}

// ---------------- K5: parallel suppression bit-matrix (j>i & IoU>T) ----------------
__global__ void __launch_bounds__(256) k_mask(const int* __restrict__ kcnt,
    const float* __restrict__ sx1, const float* __restrict__ sy1,
    const float* __restrict__ sx2, const float* __restrict__ sy2,
    const float* __restrict__ sar, unsigned* __restrict__ maskw) {
  const int img = blockIdx.y;
  const int K = kcnt[img];
  const int tid = threadIdx.x;
  const int r = blockIdx.x * 32 + (tid >> 3);     // 32 rows per block, 8 threads/row
  const float* X1 = sx1 + (size_t)img * 1024;
  const float* Y1 = sy1 + (size_t)img * 1024;
  const float* X2 = sx2 + (size_t)img * 1024;
  const float* Y2 = sy2 + (size_t)img * 1024;
  const float* AR = sar + (size_t)img * 1024;
  bool rv = (r < K);
  float ax1 = 0, ay1 = 0, ax2 = 0, ay2 = 0, aar = 0;
  if (rv) { ax1 = X1[r]; ay1 = Y1[r]; ax2 = X2[r]; ay2 = Y2[r]; aar = AR[r]; }
  unsigned* out = maskw + ((size_t)img * 1024 + r) * 32;
#pragma unroll
  for (int wi = 0; wi < 4; ++wi) {
    int w = (tid & 7) * 4 + wi;
    unsigned bits = 0u;
    if (rv) {
      int j0 = w * 32;
      for (int b = 0; b < 32; ++b) {
        int j = j0 + b;
        if (j > r && j < K) {
          float iw = fminf(ax2, X2[j]) - fmaxf(ax1, X1[j]);
          float ih = fminf(ay2, Y2[j]) - fmaxf(ay1, Y1[j]);
          float inter = fmaxf(iw, 0.f) * fmaxf(ih, 0.f);
          float uni = aar + AR[j] - inter;
          float iou = inter / fmaxf(uni, 1e-6f);
          if (iou > NMS_T) bits |= (1u << b);
        }
      }
    }
    out[w] = bits;
  }
}

// ---------------- K6: TDM mask -> 128KB LDS, wave32 serial scan, emit top-100 --------
__global__ void __launch_bounds__(1024) k_scan_out(Ptrs p,
    const int* __restrict__ kcnt, const float* __restrict__ ssc,
    const unsigned* __restrict__ sid, const unsigned* __restrict__ maskw,
    float* __restrict__ dets, float* __restrict__ ocls) {
  __shared__ unsigned m[1024 * 32];          // 128 KB — needs CDNA5's 320 KB LDS/WGP
  __shared__ volatile unsigned keepw[32];
  __shared__ int selI[POST_TOPK];
  __shared__ float selS[POST_TOPK];
  const int img = blockIdx.x;
  const int tid = threadIdx.x;
  const int K = kcnt[img];

  // one TDM descriptor moves the whole 128KB suppression matrix into LDS
  if (tid == 0) {
    const void* src = (const void*)(maskw + (size_t)img * 1024 * 32);
    tdm_load_1d((unsigned)(uintptr_t)(&m[0]), src, 1024u * 32u);
  }
  __builtin_amdgcn_s_wait_tensorcnt(0);
  __syncthreads();

  if (tid < 32) {
    unsigned kw = 0u;
    for (int b = 0; b < 32; ++b) if (tid * 32 + b < K) kw |= (1u << b);
    keepw[tid] = kw;
  }
  __syncthreads();

  // exact greedy NMS: single wave32, 1 LDS word per lane, K serial steps
  if (tid < 32) {
    for (int i = 0; i < K; ++i) {
      unsigned kv = keepw[i >> 5];
      if ((kv >> (i & 31)) & 1u) keepw[tid] &= ~m[i * 32 + tid];
    }
  }
  __syncthreads();

  if (tid == 0) {
    int c = 0;
    for (int i = 0; i < PRE_TOPK && c < POST_TOPK; ++i)
      if (i < K && ((keepw[i >> 5] >> (i & 31)) & 1u)) {
        selI[c] = i; selS[c] = ssc[(size_t)img * 1024 + i]; ++c;
      }
    for (int i = 0; i < PRE_TOPK && c < POST_TOPK; ++i) {   // -1 padding, index order
      bool kept = (i < K) && ((keepw[i >> 5] >> (i & 31)) & 1u);
      if (!kept) { selI[c] = i; selS[c] = -1.0f; ++c; }
    }
    for (; c < POST_TOPK; ++c) { selI[c] = 0; selS[c] = -1.0f; }
  }
  __syncthreads();

  if (tid < POST_TOPK) {
    int i = selI[tid];
    unsigned idx = sid[(size_t)img * 1024 + i];
    int loc = (int)idx / NCLS;
    int c = (int)idx - loc * NCLS;
    float x1, y1, x2, y2;
    decode_box(p, img, loc, x1, y1, x2, y2);   // raw (non-offset) boxes for output
    float* dr = dets + ((size_t)img * POST_TOPK + tid) * 5;
    dr[0] = x1; dr[1] = y1; dr[2] = x2; dr[3] = y2; dr[4] = selS[tid];
    ocls[(size_t)img * POST_TOPK + tid] = (float)c;
  }
}

// ---------------- host ----------------
extern "C" void kernel_launch(void* const* d_in, const int* in_sizes, int n_in,
                              void* d_out, int out_size, void* d_ws, size_t ws_size,
                              hipStream_t stream) {
  (void)in_sizes; (void)n_in; (void)out_size; (void)ws_size;
  Ptrs P;
  for (int l = 0; l < 5; ++l) {
    P.cls[l] = (const float*)d_in[3 * l + 0];
    P.box[l] = (const float*)d_in[3 * l + 1];
    P.ctr[l] = (const float*)d_in[3 * l + 2];
  }
  char* ws = (char*)d_ws;
  int*      hist   = (int*)(ws + OFF_HIST);
  int*      thresh = (int*)(ws + OFF_THRESH);
  int*      cnt    = (int*)(ws + OFF_CNT);
  int*      kcnt   = (int*)(ws + OFF_KCNT);
  uint2*    cand   = (uint2*)(ws + OFF_CAND);
  float*    ssc    = (float*)(ws + OFF_SSC);
  unsigned* sid    = (unsigned*)(ws + OFF_SID);
  float*    sx1    = (float*)(ws + OFF_SX1);
  float*    sy1    = (float*)(ws + OFF_SY1);
  float*    sx2    = (float*)(ws + OFF_SX2);
  float*    sy2    = (float*)(ws + OFF_SY2);
  float*    sar    = (float*)(ws + OFF_SAR);
  unsigned* maskw  = (unsigned*)(ws + OFF_MASK);
  float*    dets   = (float*)d_out;
  float*    ocls   = dets + (size_t)NIMG * POST_TOPK * 5;

  int initN = (int)(OFF_CAND / 4);   // hist + thresh + cnt + kcnt region
  k_init<<<(initN + 255) / 256, 256, 0, stream>>>((int*)ws, initN);

  dim3 gScan(64, NIMG);
  k_hist<<<gScan, 256, 0, stream>>>(P, hist);
  k_thresh<<<1, 32, 0, stream>>>(hist, thresh);
  k_compact<<<gScan, 256, 0, stream>>>(P, thresh, cnt, cand);
  k_sort_decode<<<NIMG, 1024, 0, stream>>>(P, cnt, cand, kcnt, ssc, sid,
                                           sx1, sy1, sx2, sy2, sar);
  dim3 gMask(32, NIMG);
  k_mask<<<gMask, 256, 0, stream>>>(kcnt, sx1, sy1, sx2, sy2, sar, maskw);
  k_scan_out<<<NIMG, 1024, 0, stream>>>(P, kcnt, ssc, sid, maskw, dets, ocls);
}